// feature_space_loss_24876450578879
// MI455X (gfx1250) — compile-verified
//
#include <hip/hip_runtime.h>
#include <hip/hip_bf16.h>

typedef float v2f __attribute__((ext_vector_type(2)));
typedef float v8f __attribute__((ext_vector_type(8)));

#define BSZ   4
#define CSZ   17
#define NSZ   4096
#define KNB   7
#define PADW  20          // 17 channels padded to 20 (5 K-chunks of 4, zero pad)
#define TSZ   289         // 17*17
#define NPTS  (BSZ*NSZ)   // 16384
#define WPB   4           // waves per block in knn kernel (split candidate dim)

// ---------------- Kernel A: per-point squared norms ----------------
__global__ __launch_bounds__(256) void sq_kernel(const float* __restrict__ logits,
                                                 float* __restrict__ sq) {
    int p = blockIdx.x * 256 + threadIdx.x;
    if (p >= NPTS) return;
    int b = p >> 12;           // p / NSZ
    int n = p & (NSZ - 1);
    const float* xb = logits + (size_t)b * CSZ * NSZ;
    float s = 0.f;
    #pragma unroll
    for (int c = 0; c < CSZ; ++c) {
        float v = xb[c * NSZ + n];
        s += v * v;
    }
    sq[p] = s;
}

// ---------------- Kernel B: WMMA Gram tiles + top-7 neighbor search ----------------
// 128 threads = 4 wave32s per block; all waves share one 16-query tile (B operand),
// each wave scans a strided quarter of the candidate tiles with
// V_WMMA_F32_16X16X4_F32 (D[m=candidate][n=query]); 8 sorted 7-lists merged at end.
__global__ __launch_bounds__(128) void knn_kernel(const float* __restrict__ logits,
                                                  const int* __restrict__ labels,
                                                  const float* __restrict__ sq,
                                                  int* __restrict__ nbrIdx,
                                                  float* __restrict__ nbrW) {
    __shared__ float ldsQ[16 * PADW];
    __shared__ float ldsX[WPB][16 * PADW];
    __shared__ float ldsSq[WPB][16];
    __shared__ float ldsD[WPB * 32 * KNB];
    __shared__ int   ldsI[WPB * 32 * KNB];

    const int wg    = blockIdx.x;        // 0..1023
    const int b     = wg >> 8;           // 256 query tiles per batch
    const int qbase = (wg & 255) * 16;
    const int tid   = threadIdx.x;
    const int wv    = tid >> 5;          // wave id 0..3
    const int lane  = tid & 31;

    const float* xb = logits + (size_t)b * CSZ * NSZ;

    // zero-init (covers K-pad channels 17..19, never rewritten)
    for (int e = tid; e < 16 * PADW; e += 128) ldsQ[e] = 0.f;
    for (int e = lane; e < 16 * PADW; e += 32) ldsX[wv][e] = 0.f;
    __syncthreads();

    // stage query tile once per block: ldsQ[n][c]
    for (int e = tid; e < 16 * CSZ; e += 128) {
        int c = e >> 4, n = e & 15;
        ldsQ[n * PADW + c] = xb[c * NSZ + qbase + n];
    }
    __syncthreads();

    const int half = lane >> 4;          // operand K-half select
    const int mr   = lane & 15;

    // B operand (queries): lane = column n=mr, VGPR pair = K {4c+2h, 4c+2h+1}
    v2f qv[5];
    #pragma unroll
    for (int c = 0; c < 5; ++c) {
        qv[c].x = ldsQ[mr * PADW + 4 * c + 2 * half];
        qv[c].y = ldsQ[mr * PADW + 4 * c + 2 * half + 1];
    }

    const int   qi  = qbase + mr;              // this lane's query point
    const float sqq = sq[b * NSZ + qi];

    float bd[KNB]; int bi[KNB];                // sorted ascending per-lane top-7
    #pragma unroll
    for (int t = 0; t < KNB; ++t) { bd[t] = 3.4e38f; bi[t] = -1; }

    // each wave scans 64 of the 256 candidate tiles (equal trip count -> safe barriers)
    for (int jb = wv * 16; jb < NSZ; jb += WPB * 16) {
        __syncthreads();
        for (int e = lane; e < 16 * CSZ; e += 32) {
            int c = e >> 4, n = e & 15;
            ldsX[wv][n * PADW + c] = xb[c * NSZ + jb + n];
        }
        if (lane < 16) ldsSq[wv][lane] = sq[b * NSZ + jb + lane];
        __syncthreads();

        // 16x16 Gram tile, K=17 (padded to 20) in 5 WMMA chunks — exact fp32
        v8f acc = {0.f, 0.f, 0.f, 0.f, 0.f, 0.f, 0.f, 0.f};
        #pragma unroll
        for (int c = 0; c < 5; ++c) {
            v2f av;
            av.x = ldsX[wv][mr * PADW + 4 * c + 2 * half];
            av.y = ldsX[wv][mr * PADW + 4 * c + 2 * half + 1];
            acc = __builtin_amdgcn_wmma_f32_16x16x4_f32(
                false, av, false, qv[c], (short)0, acc, false, false);
        }

        // contiguous 8-norm preload (vectorizable ds_load)
        float sqc[8];
        #pragma unroll
        for (int v = 0; v < 8; ++v) sqc[v] = ldsSq[wv][v + 8 * half];

        // branchless distance; divergence only on rare accept
        #pragma unroll
        for (int v = 0; v < 8; ++v) {
            int   j = jb + v + 8 * half;
            float d = sqq + sqc[v] - 2.0f * acc[v];
            d = (j == qi) ? 3.0e38f : d;               // exclude self
            if (d < bd[KNB - 1]) {
                int t = KNB - 1;
                while (t > 0 && bd[t - 1] > d) {
                    bd[t] = bd[t - 1]; bi[t] = bi[t - 1]; --t;
                }
                bd[t] = d; bi[t] = j;
            }
        }
    }

    __syncthreads();
    #pragma unroll
    for (int t = 0; t < KNB; ++t) {
        ldsD[(wv * 32 + lane) * KNB + t] = bd[t];
        ldsI[(wv * 32 + lane) * KNB + t] = bi[t];
    }
    __syncthreads();

    // threads 0..15 (wave 0): merge 8 sorted lists per query (4 waves x 2 halves)
    if (tid < 16) {
        int pos[8];
        #pragma unroll
        for (int r = 0; r < 8; ++r) pos[r] = 0;

        const int qidx  = qbase + tid;
        int       labi  = labels[b * NSZ + qidx];
        size_t    pbase = (size_t)(b * NSZ + qidx) * KNB;

        #pragma unroll
        for (int t = 0; t < KNB; ++t) {
            float best = 3.4e38f; int br = 0;
            #pragma unroll
            for (int r = 0; r < 8; ++r) {
                int list = (r >> 1) * 32 + tid + (r & 1) * 16;
                float dv = ldsD[list * KNB + pos[r]];
                if (dv < best) { best = dv; br = r; }
            }
            int list = (br >> 1) * 32 + tid + (br & 1) * 16;
            int j = ldsI[list * KNB + pos[br]];
            pos[br]++;

            int   labj = labels[b * NSZ + j];
            float sgn  = (labj == labi) ? 1.0f : -1.0f;
            nbrIdx[pbase + t] = b * NSZ + j;           // flattened, like reference idxf
            nbrW[pbase + t]   = sgn * __expf(-0.5f * best);
        }
    }
}

// ---------------- Kernel C: weighted T-distance per point ----------------
__global__ __launch_bounds__(128) void tdist_kernel(const float* __restrict__ insT,
                                                    const int* __restrict__ nbrIdx,
                                                    const float* __restrict__ nbrW,
                                                    float* __restrict__ partial) {
    const int p   = blockIdx.x;
    const int tid = threadIdx.x;
    __shared__ float Tp[TSZ];
    __shared__ float red[128];

    const float* tp = insT + (size_t)p * TSZ;
    for (int e = tid; e < TSZ; e += 128) Tp[e] = tp[e];
    __syncthreads();

    float acc = 0.f;
    #pragma unroll
    for (int t = 0; t < KNB; ++t) {
        int   j = nbrIdx[p * KNB + t];
        float w = nbrW[p * KNB + t];
        const float* tj = insT + (size_t)j * TSZ;
        float s = 0.f;
        for (int e = tid; e < TSZ; e += 128) {
            float df = Tp[e] - tj[e];
            s += df * df;
        }
        acc += w * s;
    }
    red[tid] = acc;
    __syncthreads();
    #pragma unroll
    for (int s = 64; s > 0; s >>= 1) {
        if (tid < s) red[tid] += red[tid + s];
        __syncthreads();
    }
    if (tid == 0) partial[p] = red[0];
}

// ---------------- Kernel D: deterministic final reduction / mean ----------------
__global__ __launch_bounds__(256) void reduce_kernel(const float* __restrict__ partial,
                                                     float* __restrict__ out) {
    __shared__ float red[256];
    int tid = threadIdx.x;
    float s = 0.f;
    for (int i = tid; i < NPTS; i += 256) s += partial[i];
    red[tid] = s;
    __syncthreads();
    #pragma unroll
    for (int st = 128; st > 0; st >>= 1) {
        if (tid < st) red[tid] += red[tid + st];
        __syncthreads();
    }
    if (tid == 0) out[0] = red[0] / (float)(NPTS * KNB);
}

extern "C" void kernel_launch(void* const* d_in, const int* in_sizes, int n_in,
                              void* d_out, int out_size, void* d_ws, size_t ws_size,
                              hipStream_t stream) {
    const float* logits = (const float*)d_in[0];   // (4,17,4096) f32
    const int*   labels = (const int*)  d_in[1];   // (4,4096)    i32
    const float* insT   = (const float*)d_in[2];   // (16384,17,17) f32
    float* out = (float*)d_out;

    char* ws = (char*)d_ws;
    // workspace layout: sq | nbrIdx | nbrW | partial  (~1.04 MB)
    float* sq      = (float*)ws;                                   // 16384 f
    int*   nbrIdx  = (int*)  (ws + 65536);                         // 16384*7 i32
    float* nbrW    = (float*)(ws + 65536 + 458752);                // 16384*7 f32
    float* partial = (float*)(ws + 65536 + 2 * 458752);            // 16384 f

    sq_kernel    <<<NPTS / 256, 256, 0, stream>>>(logits, sq);
    knn_kernel   <<<NPTS / 16,  128, 0, stream>>>(logits, labels, sq, nbrIdx, nbrW);
    tdist_kernel <<<NPTS,       128, 0, stream>>>(insT, nbrIdx, nbrW, partial);
    reduce_kernel<<<1,          256, 0, stream>>>(partial, out);
}